// HGNN_77644418777151
// MI455X (gfx1250) — compile-verified
//
#include <hip/hip_runtime.h>
#include <stdint.h>

typedef float v2f __attribute__((ext_vector_type(2)));
typedef float v8f __attribute__((ext_vector_type(8)));

#define D 128  // IN_DIM == HID == OUT == 128

// Wave-relative LDS byte offset of a __shared__ object: on AMDGPU the generic
// (flat) address of LDS carries the DS-usable offset in its low 32 bits.
__device__ __forceinline__ unsigned lds_off(const void* p) {
    return (unsigned)(uintptr_t)p;
}

// ---------------------------------------------------------------------------
// Dense GEMM + bias (+ optional fused ReLU on the input matrix).
//   Y[nrows, 128] = act(A[nrows,128]) @ W[128,128] + bias[128]
// One block = 256 threads = 8 waves. Block owns a 16-row strip; wave w owns
// output cols [16w, 16w+16). K-loop in steps of 4 with V_WMMA_F32_16X16X4_F32.
// W is staged into LDS with CDNA5 GLOBAL_LOAD_ASYNC_TO_LDS_B128 (ASYNCcnt),
// bypassing the VGPR round-trip.
// ---------------------------------------------------------------------------
__global__ __launch_bounds__(256) void hgnn_gemm_bias(
    const float* __restrict__ A, const float* __restrict__ W,
    const float* __restrict__ bias, float* __restrict__ Y,
    int nrows, int relu_in)
{
    __shared__ float Ws[D * D];     // 64 KB
    __shared__ float As[16 * D];    //  8 KB
    __shared__ float bs[D];

    const int tid  = threadIdx.x;
    const int row0 = blockIdx.x * 16;

    // --- Async-stage full W (64 KB) directly into LDS: 16 x b128 per thread.
    {
        const unsigned lw = lds_off(Ws);
        const char* gw = (const char*)W;
        for (int i = tid * 16; i < D * D * 4; i += 256 * 16) {
            asm volatile("global_load_async_to_lds_b128 %0, %1, off"
                         :: "v"(lw + i), "v"(gw + i)
                         : "memory");
        }
    }

    if (tid < D) bs[tid] = bias[tid];

    // --- Stage 16x128 A strip (ReLU fused in-register for layer 2).
    for (int i = tid * 4; i < 16 * D; i += 256 * 4) {
        int r = i >> 7, c = i & (D - 1);
        int gr = row0 + r;
        float4 v = make_float4(0.f, 0.f, 0.f, 0.f);
        if (gr < nrows) v = *(const float4*)(A + (size_t)gr * D + c);
        if (relu_in) {
            v.x = fmaxf(v.x, 0.f); v.y = fmaxf(v.y, 0.f);
            v.z = fmaxf(v.z, 0.f); v.w = fmaxf(v.w, 0.f);
        }
        *(float4*)(As + i) = v;
    }

    // Drain this wave's async LDS writes, then workgroup barrier.
    asm volatile("s_wait_asynccnt 0" ::: "memory");
    __syncthreads();

    const int wave = tid >> 5;
    const int lane = tid & 31;
    const int hi   = lane >> 4;   // half-wave select
    const int l    = lane & 15;
    const int col0 = wave * 16;

    v8f acc = {};
    #pragma unroll 4
    for (int k0 = 0; k0 < D; k0 += 4) {
        const int ka = k0 + 2 * hi;  // lanes 0-15: K=k0,k0+1 ; lanes 16-31: K=k0+2,k0+3
        v2f a, b;
        a.x = As[l * D + ka];
        a.y = As[l * D + ka + 1];
        b.x = Ws[ka * D + col0 + l];
        b.y = Ws[(ka + 1) * D + col0 + l];
        // 8 args: (neg_a, A, neg_b, B, c_mod, C, reuse_a, reuse_b)
        acc = __builtin_amdgcn_wmma_f32_16x16x4_f32(
            false, a, false, b, (short)0, acc, false, false);
    }

    // C/D layout: VGPR r -> M = r + 8*hi, N = l.
    const int   col  = col0 + l;
    const float bias_c = bs[col];
    float* yb = Y + (size_t)(row0 + 8 * hi) * D + col;
    if (row0 + 16 <= nrows) {            // full tile: unconditional stores
        #pragma unroll
        for (int r = 0; r < 8; ++r)
            yb[(size_t)r * D] = acc[r] + bias_c;
    } else {                             // ragged tail (not hit for N%16==0)
        #pragma unroll
        for (int r = 0; r < 8; ++r)
            if (row0 + r + 8 * hi < nrows)
                yb[(size_t)r * D] = acc[r] + bias_c;
    }
}

// ---------------------------------------------------------------------------
// COO SpMM scatter: out[rows[e], :] += vals[e] * Y[cols[e], :]
// 32 lanes per edge; each lane handles a float4 column slab. Y and out
// (51 MB each) are L2-resident on MI455X (192 MB L2), so the gather and the
// f32 atomics resolve in L2, not HBM.
// ---------------------------------------------------------------------------
__global__ __launch_bounds__(256) void hgnn_spmm_atomic(
    const int* __restrict__ rows, const int* __restrict__ cols,
    const float* __restrict__ vals, const float* __restrict__ Y,
    float* __restrict__ out, int nedges)
{
    const long long idx = (long long)blockIdx.x * blockDim.x + threadIdx.x;
    const long long e   = idx >> 5;          // wave-uniform edge id
    if (e >= nedges) return;
    const int seg = (int)(idx & 31) * 4;     // column slab

    const int   r = rows[e];
    const int   c = cols[e];
    const float v = vals[e];

    const float4 x = *(const float4*)(Y + (size_t)c * D + seg);
    float* o = out + (size_t)r * D + seg;
    atomicAdd(o + 0, v * x.x);
    atomicAdd(o + 1, v * x.y);
    atomicAdd(o + 2, v * x.z);
    atomicAdd(o + 3, v * x.w);
}

// ---------------------------------------------------------------------------
// Pipeline:  Y = X@W1+b1 ; h(out) = H@Y ; Y = relu(h)@W2+b2 ; out = H@Y
// d_out doubles as the hidden-layer accumulator (stream ordering is safe).
// ---------------------------------------------------------------------------
extern "C" void kernel_launch(void* const* d_in, const int* in_sizes, int n_in,
                              void* d_out, int out_size, void* d_ws, size_t ws_size,
                              hipStream_t stream) {
    (void)n_in; (void)out_size; (void)ws_size;
    const float* X  = (const float*)d_in[0];
    const int*   Hr = (const int*)d_in[1];
    const int*   Hc = (const int*)d_in[2];
    const float* Hv = (const float*)d_in[3];
    const float* W1 = (const float*)d_in[4];
    const float* b1 = (const float*)d_in[5];
    const float* W2 = (const float*)d_in[6];
    const float* b2 = (const float*)d_in[7];
    float* out = (float*)d_out;

    const int Nn = in_sizes[0] / D;   // 100000 nodes
    const int E_ = in_sizes[1];       // 1.6M edges
    float* Y = (float*)d_ws;          // [Nn, 128] fp32 scratch (51.2 MB)

    const int gemm_blocks = (Nn + 15) / 16;
    const int spmm_blocks = (int)(((long long)E_ * 32 + 255) / 256);
    const size_t out_bytes = (size_t)Nn * D * sizeof(float);

    // Layer 1
    hgnn_gemm_bias<<<gemm_blocks, 256, 0, stream>>>(X, W1, b1, Y, Nn, 0);
    hipMemsetAsync(out, 0, out_bytes, stream);
    hgnn_spmm_atomic<<<spmm_blocks, 256, 0, stream>>>(Hr, Hc, Hv, Y, out, E_);

    // Layer 2 (ReLU fused into GEMM's A-staging)
    hgnn_gemm_bias<<<gemm_blocks, 256, 0, stream>>>(out, W2, b2, Y, Nn, 1);
    hipMemsetAsync(out, 0, out_bytes, stream);
    hgnn_spmm_atomic<<<spmm_blocks, 256, 0, stream>>>(Hr, Hc, Hv, Y, out, E_);
}